// FullPrecisionAttentionBlockShared_87290915324059
// MI455X (gfx1250) — compile-verified
//
#include <hip/hip_runtime.h>
#include <hip/hip_bf16.h>

#define TT 512
#define DD 64
#define EPSF 1e-5f

typedef float v2f __attribute__((ext_vector_type(2)));
typedef float v8f __attribute__((ext_vector_type(8)));

__device__ __forceinline__ void get_ab(const float* __restrict__ lambda1, int d,
                                       float& a, float& b) {
  int h = d & 31;
  a = -fabsf(lambda1[h]);          // a_half replicated
  float bh = lambda1[32 + h];
  b = (d < 32) ? bh : -bh;         // b = [b_half, -b_half]
}

// ---------------- kernel 1: lambda_h + mat_exp outputs ----------------
__global__ void prep_kernel(const float* __restrict__ lambda1, const float* __restrict__ tm,
                            float* __restrict__ out_lh, float* __restrict__ out_me) {
  int d = threadIdx.x;             // 0..63
  float a, b; get_ab(lambda1, d, a, b);
  out_lh[d] = a;
  out_lh[DD + d] = b;
  float dT = tm[TT] - tm[TT - 1];
  float me = __expf(a * dT);
  out_me[d]      = me * __cosf(b * dT);
  out_me[DD + d] = me * __sinf(b * dT);
}

// ---------------- kernel 2: Q/K projection via f32 WMMA ----------------
// dst[m][n] = sum_k A[m][k] * W[n][k] + bias[n], M=1024 (c,t flat), N=K=64.
// One wave per 16x16 output tile, K stepped by 4 (v_wmma_f32_16x16x4_f32).
__global__ void proj_kernel(const float* __restrict__ Zq, const float* __restrict__ Zk,
                            const float* __restrict__ Wq, const float* __restrict__ bq,
                            const float* __restrict__ Wk, const float* __restrict__ bk,
                            float* __restrict__ Qp, float* __restrict__ Kp) {
  int bid  = blockIdx.x;           // 0..511
  int proj = bid >> 8;             // 0 = Q, 1 = K
  int rem  = bid & 255;
  int mt   = rem >> 2;             // 0..63 (M tiles of 16 over 1024 rows)
  int nt   = rem & 3;              // 0..3  (N tiles of 16 over 64 cols)
  const float* A    = proj ? Zk : Zq;
  const float* W    = proj ? Wk : Wq;
  const float* bias = proj ? bk : bq;
  float* dst        = proj ? Kp : Qp;

  int lane = threadIdx.x;          // 0..31 (wave32; EXEC all ones)
  int half = lane >> 4;            // lanes 16..31 hold K+2..K+3
  int l15  = lane & 15;
  int arow = mt * 16 + l15;        // A-matrix M index for this lane
  int ncol = nt * 16 + l15;        // B/C/D N index for this lane

  v8f acc = {};
  for (int k0 = 0; k0 < DD; k0 += 4) {
    int kb = k0 + half * 2;
    v2f av = *(const v2f*)(A + arow * DD + kb);   // A[M][kb], A[M][kb+1]
    v2f bv = *(const v2f*)(W + ncol * DD + kb);   // B[k][n] = W[n][k]
    acc = __builtin_amdgcn_wmma_f32_16x16x4_f32(false, av, false, bv,
                                                (short)0, acc, false, false);
  }
  float bb = bias[ncol];
  #pragma unroll
  for (int r = 0; r < 8; ++r) {
    int row = mt * 16 + r + half * 8;             // C/D layout: lanes 16..31 -> M+8
    dst[row * DD + ncol] = acc[r] + bb;
  }
}

// ---------------- kernel 3: dist[i][j] = |nu| * sum_d mahal ----------------
__global__ void dist_kernel(const float* __restrict__ Qp, const float* __restrict__ Kp,
                            const float* __restrict__ lambda1, const float* __restrict__ lOm,
                            const float* __restrict__ lGa, const float* __restrict__ lC,
                            const float* __restrict__ tm, const float* __restrict__ nup,
                            float* __restrict__ dist) {
  __shared__ float sQr[DD], sQi[DD], sa[DD], sb[DD], sVf[DD], sG[DD];
  int i   = blockIdx.x;
  int tid = threadIdx.x;           // 256 threads
  if (tid < DD) {
    sQr[tid] = Qp[i * DD + tid];
    sQi[tid] = Qp[(TT + i) * DD + tid];
    float a, b; get_ab(lambda1, tid, a, b);
    sa[tid] = a; sb[tid] = b;
    sVf[tid] = lC[tid] * fabsf(lOm[tid]) / (-2.f * a + EPSF);
    sG[tid]  = fabsf(lGa[tid]) + EPSF;
  }
  __syncthreads();
  float nu = fabsf(nup[0]);
  float ti = tm[i];
  for (int j = tid; j <= i; j += 256) {
    float dt  = ti - tm[j];
    float adt = fabsf(dt);
    float acc = 0.f;
    for (int d = 0; d < DD; ++d) {
      float a = sa[d], b = sb[d];
      float e  = __expf(a * dt);
      float cs = __cosf(b * dt), sn = __sinf(b * dt);
      float kr = Kp[j * DD + d];
      float ki = Kp[(TT + j) * DD + d];
      float rr = sQr[d] - e * (cs * kr - sn * ki);
      float ri = sQi[d] - e * (sn * kr + cs * ki);
      float vij = sVf[d] * (1.f - __expf(2.f * a * adt));
      acc += (rr * rr + ri * ri) / (vij + sG[d]);
    }
    dist[i * TT + j] = nu * acc;
  }
}

// ---------------- kernel 4: softmax + Zhat + Q_ij + est_latent ----------------
// Block per query i, thread per d. V_ij cached in LDS (128KB < 320KB WGP LDS).
// score = -|tau|*log(base+eps) is monotone in base, so row-max = base-min.
__global__ void attn_kernel(const float* __restrict__ Zv, const float* __restrict__ tm,
                            const float* __restrict__ lambda1, const float* __restrict__ lOm,
                            const float* __restrict__ nfp, const float* __restrict__ taup,
                            const float* __restrict__ etap, const float* __restrict__ lC,
                            const float* __restrict__ dist,
                            float* __restrict__ est, float* __restrict__ qout,
                            float* __restrict__ zhat) {
  extern __shared__ float smem[];
  float* sV    = smem;             // TT*DD  : V_ij for this i
  float* sdist = smem + TT * DD;   // TT
  float* st    = sdist + TT;       // TT
  int i = blockIdx.x;
  int d = threadIdx.x;             // 0..63

  for (int j = d; j < TT; j += DD) {
    st[j]    = tm[j];
    sdist[j] = (j <= i) ? dist[i * TT + j] : 0.f;
  }
  __syncthreads();

  float a, b; get_ab(lambda1, d, a, b);
  float vfac = lC[d] * fabsf(lOm[d]) / (-2.f * a + EPSF);
  float nfv  = fabsf(nfp[0]);
  float tauv = fabsf(taup[0]);
  float ti   = st[i];

  // pass A: V_ij into LDS, track min(base) (== max score)
  float bmin = 3.4e38f;
  for (int j = 0; j <= i; ++j) {
    float dt  = ti - st[j];
    float vij = vfac * (1.f - __expf(2.f * a * fabsf(dt)));
    sV[j * DD + d] = vij;                       // each thread owns column d
    float base = vij * (nfv + sdist[j]);
    bmin = fminf(bmin, base);
  }
  float lb = __logf(bmin + EPSF);

  // pass B: softmax denominator
  float s = 0.f;
  for (int j = 0; j <= i; ++j) {
    float base = sV[j * DD + d] * (nfv + sdist[j]);
    s += __expf(-tauv * (__logf(base + EPSF) - lb));
  }
  float inv_s = 1.f / s;

  // pass C: single streaming write of Q_ij / Zhat, fused est_v_avg
  float accr = 0.f, acci = 0.f;
  for (int j = 0; j < TT; ++j) {
    size_t o = ((size_t)i * TT + j) * DD + d;
    if (j <= i) {
      float dt   = ti - st[j];
      float base = sV[j * DD + d] * (nfv + sdist[j]);
      float q = __expf(-tauv * (__logf(base + EPSF) - lb)) * inv_s;
      float e  = __expf(a * dt);
      float cs = __cosf(b * dt), sn = __sinf(b * dt);
      float vr = Zv[j * DD + d];
      float vi = Zv[(TT + j) * DD + d];
      float zr = e * (cs * vr - sn * vi);
      float zi = e * (sn * vr + cs * vi);
      qout[o] = q;
      zhat[o] = zr;
      zhat[(size_t)TT * TT * DD + o] = zi;
      accr += q * zr;
      acci += q * zi;
    } else {                                    // masked region: explicit zeros
      qout[o] = 0.f;
      zhat[o] = 0.f;
      zhat[(size_t)TT * TT * DD + o] = 0.f;
    }
  }
  float eta = 1.f / (1.f + __expf(-etap[d]));
  float vr0 = Zv[i * DD + d], vi0 = Zv[(TT + i) * DD + d];
  est[i * DD + d]        = (1.f - eta) * vr0 + eta * accr;
  est[(TT + i) * DD + d] = (1.f - eta) * vi0 + eta * acci;
}

extern "C" void kernel_launch(void* const* d_in, const int* in_sizes, int n_in,
                              void* d_out, int out_size, void* d_ws, size_t ws_size,
                              hipStream_t stream) {
  const float* Zq  = (const float*)d_in[0];
  const float* Zk  = (const float*)d_in[1];
  const float* Zv  = (const float*)d_in[2];
  const float* tm  = (const float*)d_in[3];
  const float* Wq  = (const float*)d_in[4];
  const float* bq  = (const float*)d_in[5];
  const float* Wk  = (const float*)d_in[6];
  const float* bk  = (const float*)d_in[7];
  const float* lambda1 = (const float*)d_in[8];
  const float* lOm = (const float*)d_in[9];
  const float* lGa = (const float*)d_in[10];
  const float* nf  = (const float*)d_in[11];
  const float* tau = (const float*)d_in[12];
  const float* nu  = (const float*)d_in[13];
  const float* etp = (const float*)d_in[14];
  const float* lC  = (const float*)d_in[15];

  float* out = (float*)d_out;
  // flat output layout: est(65536) | Q_ij(16777216) | Zhat(33554432) | lambda_h(128) | mat_exp(128)
  float* out_est  = out;
  float* out_q    = out + 65536;
  float* out_zhat = out + 16842752;
  float* out_lh   = out + 50397184;
  float* out_me   = out + 50397312;

  float* Qp   = (float*)d_ws;            // 1024*64
  float* Kp   = Qp + 2 * TT * DD;        // 1024*64
  float* dist = Kp + 2 * TT * DD;        // 512*512

  prep_kernel<<<1, 64, 0, stream>>>(lambda1, tm, out_lh, out_me);
  proj_kernel<<<512, 32, 0, stream>>>(Zq, Zk, Wq, bq, Wk, bk, Qp, Kp);
  dist_kernel<<<TT, 256, 0, stream>>>(Qp, Kp, lambda1, lOm, lGa, lC, tm, nu, dist);
  size_t shmem = (size_t)(TT * DD + 2 * TT) * sizeof(float);   // 132 KB dynamic LDS
  attn_kernel<<<TT, 64, shmem, stream>>>(Zv, tm, lambda1, lOm, nf, tau, etp, lC, dist,
                                         out_est, out_q, out_zhat);
}